// TemporalGNN_43044162240887
// MI455X (gfx1250) — compile-verified
//
#include <hip/hip_runtime.h>
#include <hip/hip_bf16.h>
#include <math.h>

// Problem constants (fixed by the reference)
#define NN   15      // nodes == GRU hidden size
#define HH   16384   // hidden channels == GRU sequence length
#define G3   45      // 3 * HID gates
#define GIP  48      // padded gate stride (3 x 16 N-tiles)

typedef float v2f __attribute__((ext_vector_type(2)));
typedef float v8f __attribute__((ext_vector_type(8)));

__device__ __forceinline__ float sigm_(float x) { return 1.0f / (1.0f + __expf(-x)); }
__device__ __forceinline__ float tanh_(float x) { return 2.0f / (1.0f + __expf(-2.0f * x)) - 1.0f; }

// ---------------------------------------------------------------------------
// Kernel 1: GCN aggregation collapsed to a 15-vector.
//   A[d] = dinv[d]^2 * x[d] + sum_{e: dst==d} dinv[src]*dinv[d]*x[src]
// ---------------------------------------------------------------------------
__global__ void gcn_agg_kernel(const float* __restrict__ x,
                               const int* __restrict__ ei,  // [2,E] flat
                               int E,
                               float* __restrict__ Avec) {
  __shared__ float sdinv[NN];
  const int t = threadIdx.x;
  const int* src = ei;
  const int* dst = ei + E;
  if (t < NN) {
    int cnt = 1;  // self loop
    for (int e = 0; e < E; ++e) cnt += (dst[e] == t) ? 1 : 0;
    sdinv[t] = rsqrtf((float)cnt);
  }
  __syncthreads();
  if (t < NN) {
    const float di = sdinv[t];
    float acc = di * di * x[t];
    for (int e = 0; e < E; ++e) {
      if (dst[e] == t) {
        const int s = src[e];
        acc += sdinv[s] * di * x[s];
      }
    }
    Avec[t] = acc;
  }
}

// ---------------------------------------------------------------------------
// Kernel 2: GI = relu(A ⊗ W_gcn + b_gcn)^T @ W_ih^T + b_ih  via WMMA f32 16x16x4.
//   GI[c,g] = sum_d S[c,d] * W_ih[g,d] + b_ih[g],
//   S[c,d]  = relu(A[d]*W_gcn[c] + b_gcn[c])   (generated in-register, never stored)
// One wave per 16-row M tile; 3 N-tiles (g), 4 K-steps (d, padded 15->16).
// All padding is branchless (clamped in-bounds loads + v_cndmask), keeping a
// full EXEC mask throughout — required for WMMA and avoids exec-save regions.
// ---------------------------------------------------------------------------
__global__ void gi_gemm_kernel(const float* __restrict__ Wg,
                               const float* __restrict__ bg,
                               const float* __restrict__ Wih,   // [45,15]
                               const float* __restrict__ bih,   // [45]
                               const float* __restrict__ Avec,  // [15]
                               float* __restrict__ GI) {        // [HH, GIP]
  const int wave  = blockIdx.x * (blockDim.x >> 5) + (threadIdx.x >> 5);
  const int lane  = threadIdx.x & 31;
  const int c0    = wave * 16;
  const int m     = lane & 15;     // A row / B,D column-lane
  const int khalf = lane >> 4;     // lane-half selects K pair / M half

  const float w  = Wg[c0 + m];
  const float bb = bg[c0 + m];

  // Hoist all 12 loop-invariant B fragments (W_ih^T), branchless zero-padding.
  v2f bf[4][3];
#pragma unroll
  for (int kk = 0; kk < 4; ++kk) {
#pragma unroll
    for (int t = 0; t < 3; ++t) {
#pragma unroll
      for (int v = 0; v < 2; ++v) {
        const int d  = kk * 4 + v + 2 * khalf;       // K index (pad d==15)
        const int g  = t * 16 + m;                   // N index (pad g>=45)
        const int dc = (d < NN) ? d : 0;             // clamped, always in-bounds
        const int gc = (g < G3) ? g : 0;
        const float val = Wih[gc * NN + dc];         // unconditional load
        bf[kk][t][v] = (d < NN && g < G3) ? val : 0.0f;  // v_cndmask, no branch
      }
    }
  }

  v8f acc0 = {}; v8f acc1 = {}; v8f acc2 = {};

#pragma unroll
  for (int kk = 0; kk < 4; ++kk) {
    // A fragment: a[v] = S[c0+m, 4*kk + v + 2*khalf], generated in-register.
    v2f a;
#pragma unroll
    for (int v = 0; v < 2; ++v) {
      const int d  = kk * 4 + v + 2 * khalf;
      const int dc = (d < NN) ? d : 0;
      float s = fmaf(Avec[dc], w, bb);
      s = fmaxf(s, 0.0f);                            // relu
      a[v] = (d < NN) ? s : 0.0f;
    }
    acc0 = __builtin_amdgcn_wmma_f32_16x16x4_f32(false, a, false, bf[kk][0], (short)0, acc0, false, false);
    acc1 = __builtin_amdgcn_wmma_f32_16x16x4_f32(false, a, false, bf[kk][1], (short)0, acc1, false, false);
    acc2 = __builtin_amdgcn_wmma_f32_16x16x4_f32(false, a, false, bf[kk][2], (short)0, acc2, false, false);
  }

  // D layout: lane&15 = N, M = r + 8*khalf. Add b_ih while storing.
  // bi2 load is clamped+selected (no exec-mask branch, unlike round 2).
  const float bi0  = bih[m];                           // g = m      < 45 always
  const float bi1  = bih[16 + m];                      // g = 16+m   < 45 always
  const float bi2v = bih[(m < 13) ? (32 + m) : 0];     // unconditional, in-bounds
  const float bi2  = (m < 13) ? bi2v : 0.0f;           // v_cndmask
#pragma unroll
  for (int r = 0; r < 8; ++r) {
    const int row = c0 + r + 8 * khalf;
    float* __restrict__ p = GI + (size_t)row * GIP;
    p[m]      = acc0[r] + bi0;
    p[16 + m] = acc1[r] + bi1;
    p[32 + m] = acc2[r] + bi2;   // padded columns 45..47 hold zero; never read
  }
}

// ---------------------------------------------------------------------------
// Kernel 3: sequential GRU scan (16384 steps) + fused Linear(H,1).
// Single wave32; lane i<15 owns gate triple i; W_hh rows live in registers;
// replicated h rebuilt via __shfl each step; ys never materialized.
// ---------------------------------------------------------------------------
__global__ void __launch_bounds__(32)
gru_scan_kernel(const float* __restrict__ GI,
                const float* __restrict__ Whh,   // [45,15]
                const float* __restrict__ bhh,   // [45]
                const float* __restrict__ Wlin,  // [HH]
                const float* __restrict__ blin,  // [1]
                float* __restrict__ out) {       // [15]
  const int lane  = threadIdx.x;
  const bool valid = lane < NN;
  const int i = valid ? lane : 0;

  float whr[NN], whz[NN], whn[NN];
#pragma unroll
  for (int j = 0; j < NN; ++j) {
    whr[j] = Whh[i * NN + j];
    whz[j] = Whh[(NN + i) * NN + j];
    whn[j] = Whh[(2 * NN + i) * NN + j];
  }
  const float bhr = bhh[i];
  const float bhz = bhh[NN + i];
  const float bhn = bhh[2 * NN + i];

  float h[NN];
#pragma unroll
  for (int j = 0; j < NN; ++j) h[j] = 0.0f;
  float own = 0.0f;   // this lane's previous hidden value (== h[i])
  float acc = 0.0f;   // fused final-linear accumulator

  for (int c = 0; c < HH; ++c) {
    const float* __restrict__ gi = GI + (size_t)c * GIP;
    // stream-ahead (clamped address -> no per-step branch); 32 lanes * 2 floats
    // cover the 192B row; GI is L2-resident, this hides the L2 hit latency.
    const int cp = (c + 24 < HH) ? (c + 24) : (HH - 1);
    __builtin_prefetch(GI + (size_t)cp * GIP + lane * 2, 0, 0);

    const float gir = gi[i];
    const float giz = gi[NN + i];
    const float gin = gi[2 * NN + i];

    float sr = bhr, sz = bhz, sn = bhn;
#pragma unroll
    for (int j = 0; j < NN; ++j) {
      sr = fmaf(h[j], whr[j], sr);
      sz = fmaf(h[j], whz[j], sz);
      sn = fmaf(h[j], whn[j], sn);
    }
    const float r  = sigm_(gir + sr);
    const float z  = sigm_(giz + sz);
    const float nn = tanh_(gin + r * sn);
    const float hnew = (1.0f - z) * nn + z * own;
    own = hnew;
    acc = fmaf(hnew, Wlin[c], acc);
#pragma unroll
    for (int j = 0; j < NN; ++j) h[j] = __shfl(hnew, j, 32);
  }
  if (valid) out[lane] = acc + blin[0];
}

// ---------------------------------------------------------------------------
extern "C" void kernel_launch(void* const* d_in, const int* in_sizes, int n_in,
                              void* d_out, int out_size, void* d_ws, size_t ws_size,
                              hipStream_t stream) {
  const float* x    = (const float*)d_in[0];
  const int*   ei   = (const int*)d_in[1];
  const float* Wg   = (const float*)d_in[2];
  const float* bg   = (const float*)d_in[3];
  const float* Wih  = (const float*)d_in[4];
  const float* Whh  = (const float*)d_in[5];
  const float* bih  = (const float*)d_in[6];
  const float* bhh  = (const float*)d_in[7];
  const float* Wlin = (const float*)d_in[8];
  const float* blin = (const float*)d_in[9];
  float* out = (float*)d_out;

  const int E = in_sizes[1] / 2;

  float* Avec = (float*)d_ws;                 // 15 floats (padded to 64 for alignment)
  float* GI   = (float*)d_ws + 64;            // [HH, GIP] = 3 MB, lives in L2

  gcn_agg_kernel<<<1, 32, 0, stream>>>(x, ei, E, Avec);

  // HH/16 = 1024 M-tiles, one wave each: 256 blocks * 4 waves
  gi_gemm_kernel<<<256, 128, 0, stream>>>(Wg, bg, Wih, bih, Avec, GI);

  gru_scan_kernel<<<1, 32, 0, stream>>>(GI, Whh, bhh, Wlin, blin, out);
}